// EfmLSTM_9723805958785
// MI455X (gfx1250) — compile-verified
//
#include <hip/hip_runtime.h>
#include <hip/hip_bf16.h>
#include <math.h>

typedef __attribute__((ext_vector_type(2))) float v2f;
typedef __attribute__((ext_vector_type(8))) float v8f;

#define B_   64
#define T_   512
#define D_   256
#define U_   512
#define KTOT 768            // 512 (h part) + 256 (x part)
#define NWG  32             // one workgroup per 16-unit output column tile
#define BLOCK 128           // 4 waves of 32; each wave owns a 16-row batch tile

// LDS weight layout: [kchunk(192)][gate(3)][n(16)][kk(4)]
//  -> the 3 gate fragments for one k-chunk are +0 / +64 / +128 floats
//     (256/512 bytes: both fit the 16-bit DS immediate offset field)
#define KCSTRIDE (3 * 64)   // floats per k-chunk

__global__ void efm_init_kernel(unsigned* bar) {
    if (threadIdx.x == 0) { bar[0] = 0u; bar[1] = 0u; }
}

__global__ __launch_bounds__(BLOCK) void efm_lstm_persistent(
    const float* __restrict__ inputs,            // (B, T, D)
    const float* __restrict__ input_kernel,      // (D, 3U)
    const float* __restrict__ recurrent_kernel,  // (U, 3U)
    const float* __restrict__ bias,              // (4U)
    float* __restrict__ out,                     // (B, T, U)  also serves as h history
    unsigned* __restrict__ bar_count,
    unsigned* __restrict__ bar_gen)
{
    __shared__ float wlds[192 * KCSTRIDE];       // 147456 B of the 320 KB WGP LDS

    const int jn    = blockIdx.x;                // 0..31 : unit column tile
    const int jcol0 = jn * 16;
    const int lane  = threadIdx.x & 31;
    const int wv    = threadIdx.x >> 5;          // 0..3  : batch row tile
    const int m0    = wv * 16;

    // ---- Stage the 3 gate weight column-slices (K=768 x 16) into LDS once ----
    // k < 512 -> recurrent_kernel[k][g*U + jcol0 + n], else input_kernel[k-512][...]
    for (int idx = threadIdx.x; idx < 3 * KTOT * 16; idx += BLOCK) {
        int g = idx / (KTOT * 16);
        int r = idx % (KTOT * 16);
        int k = r / 16;
        int n = r % 16;
        float w = (k < U_)
            ? recurrent_kernel[(size_t)k * (3 * U_) + g * U_ + jcol0 + n]
            : input_kernel[(size_t)(k - U_) * (3 * U_) + g * U_ + jcol0 + n];
        wlds[(k >> 2) * KCSTRIDE + g * 64 + n * 4 + (k & 3)] = w;
    }
    __syncthreads();

    // ---- Per-lane constants (each lane owns one unit column j forever) ----
    const int   n     = lane & 15;
    const int   khalf = lane >> 4;               // 0/1 : which K pair of the A/B frag
    const int   j     = jcol0 + n;
    const float b_i   = bias[j];
    const float b_c   = bias[2 * U_ + j];
    const float b_o   = bias[3 * U_ + j];
    // dead signature path: sig0 == 0  =>  f_t = sigmoid(b_f) (constant per unit)
    const float f_t   = 1.0f / (1.0f + __expf(-bias[U_ + j]));

    // Per-lane LDS fragment bases.  The X-part base offset (128 k-chunks =
    // 0x18000 bytes) is laundered through an empty asm so LLVM folds it into
    // the base address register once, instead of re-materializing it as
    // per-load VALU adds (it exceeds the 16-bit DS immediate offset field).
    const float* wlds_h = &wlds[n * 4 + khalf * 2];
    unsigned xbase = 128 * KCSTRIDE;
    asm volatile("" : "+v"(xbase));              // opaque to re-association
    const float* wlds_x = &wlds[xbase + n * 4 + khalf * 2];

    // A-fragment row this lane loads (row of the 16x4 tile = lane&15)
    const int    am      = m0 + (lane & 15);
    const float* xrow_b  = inputs + (size_t)am * T_ * D_;
    const size_t outrow  = (size_t)am * T_ * U_;

    float creg[8];
    #pragma unroll
    for (int v = 0; v < 8; ++v) creg[v] = 0.0f;

    for (int t = 0; t < T_; ++t) {
        v8f acc_i = {}; v8f acc_c = {}; v8f acc_o = {};

        // Prefetch next step's x row (inputs streamed once; hide HBM latency
        // behind this step's WMMA chain).  1 KB row -> 8 x 128B lines.
        if (t + 1 < T_) {
            const float* xn = xrow_b + (size_t)(t + 1) * D_;
            #pragma unroll
            for (int off = 0; off < D_; off += 32)
                __builtin_prefetch(xn + off, 0, 3);
        }

        // ---- K = 0..511 : h_{t-1} @ Wr  (skip at t==0, h0 == 0) ----
        if (t > 0) {
            const float* hrow = out + outrow + (size_t)(t - 1) * U_ + khalf * 2;
            #pragma unroll 4
            for (int kc = 0; kc < 128; ++kc) {
                v2f a = *(const v2f*)(hrow + kc * 4);
                const float* wb = wlds_h + kc * KCSTRIDE;
                v2f bi = *(const v2f*)(wb);
                v2f bc = *(const v2f*)(wb + 64);
                v2f bo = *(const v2f*)(wb + 128);
                acc_i = __builtin_amdgcn_wmma_f32_16x16x4_f32(false, a, false, bi, (short)0, acc_i, false, false);
                acc_c = __builtin_amdgcn_wmma_f32_16x16x4_f32(false, a, false, bc, (short)0, acc_c, false, false);
                acc_o = __builtin_amdgcn_wmma_f32_16x16x4_f32(false, a, false, bo, (short)0, acc_o, false, false);
            }
        }

        // ---- K = 512..767 : x_t @ Wi (fused input GEMM, no xk intermediate) ----
        {
            const float* xrow = xrow_b + (size_t)t * D_ + khalf * 2;
            #pragma unroll 4
            for (int kc = 0; kc < 64; ++kc) {
                v2f a = *(const v2f*)(xrow + kc * 4);
                const float* wb = wlds_x + kc * KCSTRIDE;
                v2f bi = *(const v2f*)(wb);
                v2f bc = *(const v2f*)(wb + 64);
                v2f bo = *(const v2f*)(wb + 128);
                acc_i = __builtin_amdgcn_wmma_f32_16x16x4_f32(false, a, false, bi, (short)0, acc_i, false, false);
                acc_c = __builtin_amdgcn_wmma_f32_16x16x4_f32(false, a, false, bc, (short)0, acc_c, false, false);
                acc_o = __builtin_amdgcn_wmma_f32_16x16x4_f32(false, a, false, bo, (short)0, acc_o, false, false);
            }
        }

        // ---- Gate nonlinearities + cell update; c stays in registers ----
        #pragma unroll
        for (int v = 0; v < 8; ++v) {
            int   bm = m0 + v + 8 * khalf;       // batch row of C/D element v
            float it = 1.0f / (1.0f + __expf(-(acc_i[v] + b_i)));
            float ct = tanhf(acc_c[v] + b_c);
            float ot = 1.0f / (1.0f + __expf(-(acc_o[v] + b_o)));
            float cn = f_t * creg[v] + it * ct;
            creg[v] = cn;
            out[(size_t)bm * T_ * U_ + (size_t)t * U_ + j] = ot * tanhf(cn);
        }

        // ---- Device-wide barrier with release/acquire semantics ----
        __syncthreads();
        __threadfence();                          // release: h_t stores -> device scope
        if (threadIdx.x == 0) {
            unsigned g = __atomic_load_n(bar_gen, __ATOMIC_RELAXED);
            if (atomicAdd(bar_count, 1u) == NWG - 1) {
                __atomic_store_n(bar_count, 0u, __ATOMIC_RELAXED);
                __atomic_fetch_add(bar_gen, 1u, __ATOMIC_RELEASE);
            } else {
                while (__atomic_load_n(bar_gen, __ATOMIC_ACQUIRE) == g) {
                    __builtin_amdgcn_s_sleep(2);
                }
            }
        }
        __syncthreads();
        __threadfence();                          // acquire: invalidate stale h lines
    }
}

extern "C" void kernel_launch(void* const* d_in, const int* in_sizes, int n_in,
                              void* d_out, int out_size, void* d_ws, size_t ws_size,
                              hipStream_t stream) {
    (void)in_sizes; (void)n_in; (void)out_size; (void)ws_size;
    const float* inputs           = (const float*)d_in[0];
    // d_in[1] = w_sig        : unused (signature path is structurally zero)
    const float* input_kernel     = (const float*)d_in[2];
    const float* recurrent_kernel = (const float*)d_in[3];
    // d_in[4] = forget_kernel: unused (sig0 == 0)
    const float* bias             = (const float*)d_in[5];
    float*       out              = (float*)d_out;
    unsigned*    bar              = (unsigned*)d_ws;   // [count, generation]

    efm_init_kernel<<<1, 32, 0, stream>>>(bar);
    efm_lstm_persistent<<<NWG, BLOCK, 0, stream>>>(
        inputs, input_kernel, recurrent_kernel, bias, out, bar, bar + 1);
}